// GraphEncoder_89532888252581
// MI455X (gfx1250) — compile-verified
//
#include <hip/hip_runtime.h>
#include <math.h>

// Problem constants (from reference)
#define NN 100000     // nodes
#define DD 64         // feature dim
#define EE 1200000    // edges

typedef __attribute__((ext_vector_type(2))) float v2f;
typedef __attribute__((ext_vector_type(8))) float v8f;

// ---------------- utility kernels ----------------
__global__ void fill_f32(float* __restrict__ p, float v, int n) {
    int i = blockIdx.x * blockDim.x + threadIdx.x;
    if (i < n) p[i] = v;
}

__global__ void deg_kernel(const long long* __restrict__ col, float* __restrict__ deg, int e) {
    int i = blockIdx.x * blockDim.x + threadIdx.x;
    if (i < e) atomicAdd(&deg[(int)col[i]], 1.0f);
}

__global__ void dinv_kernel(const float* __restrict__ deg, float* __restrict__ dinv, int n) {
    int i = blockIdx.x * blockDim.x + threadIdx.x;
    if (i < n) {
        float d = deg[i];
        dinv[i] = (d > 0.0f) ? rsqrtf(fmaxf(d, 1.0f)) : 0.0f;
    }
}

__global__ void norm_kernel(const long long* __restrict__ row, const long long* __restrict__ col,
                            const float* __restrict__ dinv, float* __restrict__ nrm, int e) {
    int i = blockIdx.x * blockDim.x + threadIdx.x;
    if (i < e) nrm[i] = dinv[(int)row[i]] * dinv[(int)col[i]];
}

// result = x (residual base); sum-of-embeddings accumulator = 0
__global__ void init_kernel(const float* __restrict__ x, float* __restrict__ result,
                            float* __restrict__ sumemb, int n) {
    int i = blockIdx.x * blockDim.x + threadIdx.x;
    if (i < n) { result[i] = x[i]; sumemb[i] = 0.0f; }
}

// ---------------- GEMM: out[N x 64] = in[N x 64] @ W[64 x 64] via V_WMMA_F32_16X16X4_F32 ----------------
// 256 threads = 8 waves; wave w handles 16 rows; block handles 128 rows.
// W is staged into LDS PRE-PAIRED along K: Bp[(k/2)*64 + n] = {W[k][n], W[k+1][n]},
// so each B fragment is one aligned ds_load_b64 straight into an even VGPR pair
// (no v_dual_mov shuffles before the WMMA).
// Per wave: 4 f32 accumulators (16x16) covering all 64 output cols; K-loop 16 x K=4 -> 64 v_wmma.
__global__ void __launch_bounds__(256)
gemm_wmma_kernel(const float* __restrict__ in, const float* __restrict__ W,
                 float* __restrict__ out, int nrows) {
    // 32 K-pairs x 64 cols of float2 = 16 KB
    __shared__ float2 Bp[(DD / 2) * DD];

    // cooperative stage: 512 float4-row-groups; thread t handles groups t and t+256.
    // group g: kpair = g>>4 (0..31), n4 = (g&15)*4 ; pack rows 2*kpair and 2*kpair+1.
    {
        const int t = threadIdx.x;
        #pragma unroll
        for (int j = 0; j < 2; ++j) {
            const int g  = t + j * 256;
            const int kp = g >> 4;
            const int n4 = (g & 15) << 2;
            const float4 r0 = *(const float4*)(W + (kp * 2)     * DD + n4);
            const float4 r1 = *(const float4*)(W + (kp * 2 + 1) * DD + n4);
            float2* dst = &Bp[kp * DD + n4];
            dst[0] = make_float2(r0.x, r1.x);
            dst[1] = make_float2(r0.y, r1.y);
            dst[2] = make_float2(r0.z, r1.z);
            dst[3] = make_float2(r0.w, r1.w);
        }
    }
    __syncthreads();

    const int wave   = threadIdx.x >> 5;
    const int lane   = threadIdx.x & 31;
    const int lane15 = lane & 15;
    const int hi     = lane >> 4;              // 0: lanes 0-15, 1: lanes 16-31
    const int rowTile = blockIdx.x * 128 + wave * 16;

    // A fragment row (clamped so EXEC stays all-1s for WMMA; stores are masked instead)
    int aRow = rowTile + lane15;
    if (aRow > nrows - 1) aRow = nrows - 1;
    const float* arow = in + (long long)aRow * DD;

    v8f acc0 = {}, acc1 = {}, acc2 = {}, acc3 = {};

    for (int k0 = 0; k0 < DD; k0 += 4) {
        // A 16x4 layout: lanes 0-15 hold K = k0+{0,1}; lanes 16-31 hold K = k0+{2,3}
        // contiguous pair -> single global_load_b64
        const v2f a = *(const v2f*)(arow + k0 + hi * 2);

        // B 4x16 layout: VGPR0 = row K, VGPR1 = row K+1; lanes 16-31 take K+2/K+3.
        // pre-paired in LDS -> single ds_load_b64 per fragment
        const float2* brow = &Bp[((k0 >> 1) + hi) * DD + lane15];
        const v2f b0 = *(const v2f*)(brow + 0);
        const v2f b1 = *(const v2f*)(brow + 16);
        const v2f b2 = *(const v2f*)(brow + 32);
        const v2f b3 = *(const v2f*)(brow + 48);

        acc0 = __builtin_amdgcn_wmma_f32_16x16x4_f32(false, a, false, b0, (short)0, acc0, false, false);
        acc1 = __builtin_amdgcn_wmma_f32_16x16x4_f32(false, a, false, b1, (short)0, acc1, false, false);
        acc2 = __builtin_amdgcn_wmma_f32_16x16x4_f32(false, a, false, b2, (short)0, acc2, false, false);
        acc3 = __builtin_amdgcn_wmma_f32_16x16x4_f32(false, a, false, b3, (short)0, acc3, false, false);
    }

    // C/D layout: VGPR v, lane l -> (M = v + 8*(l>=16), N = l%16)
    #pragma unroll
    for (int v = 0; v < 8; ++v) {
        int r = rowTile + v + hi * 8;
        if (r < nrows) {
            float* orow = out + (long long)r * DD + lane15;
            orow[0]  = acc0[v];
            orow[16] = acc1[v];
            orow[32] = acc2[v];
            orow[48] = acc3[v];
        }
    }
}

// ---------------- scatter: o[col] += t[row] * norm  (float4 gather, 4x f32 atomics) ----------------
__global__ void scatter_kernel(const long long* __restrict__ row, const long long* __restrict__ col,
                               const float* __restrict__ nrm, const float* __restrict__ t,
                               float* __restrict__ o, int e) {
    int idx = blockIdx.x * blockDim.x + threadIdx.x;   // e * 16 threads; 4 floats each
    if (idx >= e * 16) return;
    const int ed   = idx >> 4;
    const int part = (idx & 15) << 2;
    const int r = (int)row[ed];
    const int c = (int)col[ed];
    const float nv = nrm[ed];
    const float4 m = *(const float4*)(t + (long long)r * DD + part);
    float* dst = o + (long long)c * DD + part;
    atomicAdd(dst + 0, m.x * nv);
    atomicAdd(dst + 1, m.y * nv);
    atomicAdd(dst + 2, m.z * nv);
    atomicAdd(dst + 3, m.w * nv);
}

// ---------------- fused epilogue: +bias, mean-accum, optional cl copy, L2-normalize, residual ----------------
// wave32: one wave per row, each lane owns 2 features (float2); cross-lane reduce via shfl_xor.
__global__ void __launch_bounds__(256)
epilogue_kernel(const float* __restrict__ o, const float* __restrict__ bias,
                float* __restrict__ sumemb, float* __restrict__ result,
                float* __restrict__ cl, float* __restrict__ hout,
                float inv_layer, int write_cl, int n) {
    const int wave = threadIdx.x >> 5;
    const int lane = threadIdx.x & 31;
    const int rowI = blockIdx.x * 8 + wave;
    if (rowI >= n) return;
    const long long base = (long long)rowI * DD + lane * 2;

    float2 v = *(const float2*)(o + base);
    const float2 bb = *(const float2*)(bias + lane * 2);
    v.x += bb.x; v.y += bb.y;                       // h_layer = scatter + b

    // mean over 3 layers accumulated as sum/3
    float2 s = *(float2*)(sumemb + base);
    s.x += v.x * (1.0f / 3.0f);
    s.y += v.y * (1.0f / 3.0f);
    *(float2*)(sumemb + base) = s;

    if (write_cl) *(float2*)(cl + base) = v;        // all_embeddings_cl (layer 3, pre-norm)

    // L2 norm across the 64 features of this row (32 lanes x 2)
    float ss = v.x * v.x + v.y * v.y;
    #pragma unroll
    for (int m = 16; m >= 1; m >>= 1) ss += __shfl_xor(ss, m, 32);
    const float inv = 1.0f / fmaxf(sqrtf(ss), 1e-12f);

    float2 hn = { v.x * inv, v.y * inv };
    *(float2*)(hout + base) = hn;                   // input to next layer

    float2 r = *(float2*)(result + base);
    r.x += hn.x * inv_layer;
    r.y += hn.y * inv_layer;
    *(float2*)(result + base) = r;
}

// ---------------- host launcher ----------------
extern "C" void kernel_launch(void* const* d_in, const int* in_sizes, int n_in,
                              void* d_out, int out_size, void* d_ws, size_t ws_size,
                              hipStream_t stream) {
    (void)in_sizes; (void)n_in; (void)out_size; (void)ws_size;

    const float* x  = (const float*)d_in[0];
    const float* W[3] = { (const float*)d_in[1], (const float*)d_in[3], (const float*)d_in[5] };
    const float* b[3] = { (const float*)d_in[2], (const float*)d_in[4], (const float*)d_in[6] };
    const long long* ei  = (const long long*)d_in[7];   // int64 per reference
    const long long* row = ei;
    const long long* col = ei + EE;

    float* out_result = (float*)d_out;                      // [N, D]
    float* out_mean   = out_result + (long long)NN * DD;    // [N, D]
    float* out_cl     = out_mean   + (long long)NN * DD;    // [N, D]

    // workspace layout
    float* ws   = (float*)d_ws;
    float* deg  = ws;  ws += NN;
    float* dinv = ws;  ws += NN;
    float* nrm  = ws;  ws += EE;
    float* h    = ws;  ws += (long long)NN * DD;   // normalized layer input
    float* t    = ws;  ws += (long long)NN * DD;   // GEMM output
    float* o    = ws;  ws += (long long)NN * DD;   // scatter target

    const int ND = NN * DD;

    // degree -> dinv -> per-edge norm
    fill_f32<<<(NN + 255) / 256, 256, 0, stream>>>(deg, 0.0f, NN);
    deg_kernel<<<(EE + 255) / 256, 256, 0, stream>>>(col, deg, EE);
    dinv_kernel<<<(NN + 255) / 256, 256, 0, stream>>>(deg, dinv, NN);
    norm_kernel<<<(EE + 255) / 256, 256, 0, stream>>>(row, col, dinv, nrm, EE);

    // result = x, sum-embedding accumulator = 0
    init_kernel<<<(ND + 255) / 256, 256, 0, stream>>>(x, out_result, out_mean, ND);

    const int gemm_blocks    = (NN + 127) / 128;       // 782
    const int scatter_blocks = (EE * 16 + 255) / 256;  // 75000
    const int epi_blocks     = (NN + 7) / 8;           // 12500

    const float* cur = x;
    for (int layer = 0; layer < 3; ++layer) {
        gemm_wmma_kernel<<<gemm_blocks, 256, 0, stream>>>(cur, W[layer], t, NN);
        fill_f32<<<(ND + 255) / 256, 256, 0, stream>>>(o, 0.0f, ND);
        scatter_kernel<<<scatter_blocks, 256, 0, stream>>>(row, col, nrm, t, o, EE);
        epilogue_kernel<<<epi_blocks, 256, 0, stream>>>(
            o, b[layer], out_mean, out_result, out_cl, h,
            1.0f / (float)(layer + 1), (layer == 2) ? 1 : 0, NN);
        cur = h;
    }
}